// ExpertDecoder_12661563589300
// MI455X (gfx1250) — compile-verified
//
#include <hip/hip_runtime.h>

typedef __attribute__((ext_vector_type(2))) float v2f;
typedef __attribute__((ext_vector_type(8))) float v8f;

#define TPTS   512
#define BATCH  4096
#define BLOCK  32     // one wave32 per workgroup -> 128 WGs spread across WGPs
#define SUB    8

// ---- SEIRM dynamics: k = f(y) ------------------------------------------
__device__ __forceinline__ void seirm(const float y[5], float beta, float alpha,
                                      float gamma, float mu, float gm,
                                      float k[5]) {
    float ninf = beta * y[0] * y[2];
    float aE   = alpha * y[1];
    k[0] = -ninf;
    k[1] = ninf - aE;
    k[2] = fmaf(-gm, y[2], aE);
    k[3] = gamma * y[2];
    k[4] = mu * y[2];
}

// ---- One Dormand-Prince RK45 step with h-prescaled tableau --------------
// hc[0..19] = h * {a21, a31,a32, a41..a43, a51..a54, a61..a65, b1,b3,b4,b5,b6}
__device__ __forceinline__ void dopri5_step(float y[5], const float hc[20],
                                            float beta, float alpha,
                                            float gamma, float mu, float gm) {
    float k1[5], k2[5], k3[5], k4[5], k5[5], k6[5], yt[5];

    seirm(y, beta, alpha, gamma, mu, gm, k1);
#pragma unroll
    for (int z = 0; z < 5; ++z)
        yt[z] = fmaf(hc[0], k1[z], y[z]);
    seirm(yt, beta, alpha, gamma, mu, gm, k2);
#pragma unroll
    for (int z = 0; z < 5; ++z)
        yt[z] = fmaf(hc[2], k2[z], fmaf(hc[1], k1[z], y[z]));
    seirm(yt, beta, alpha, gamma, mu, gm, k3);
#pragma unroll
    for (int z = 0; z < 5; ++z)
        yt[z] = fmaf(hc[5], k3[z],
                     fmaf(hc[4], k2[z], fmaf(hc[3], k1[z], y[z])));
    seirm(yt, beta, alpha, gamma, mu, gm, k4);
#pragma unroll
    for (int z = 0; z < 5; ++z)
        yt[z] = fmaf(hc[9], k4[z],
                     fmaf(hc[8], k3[z],
                          fmaf(hc[7], k2[z], fmaf(hc[6], k1[z], y[z]))));
    seirm(yt, beta, alpha, gamma, mu, gm, k5);
#pragma unroll
    for (int z = 0; z < 5; ++z)
        yt[z] = fmaf(hc[14], k5[z],
                     fmaf(hc[13], k4[z],
                          fmaf(hc[12], k3[z],
                               fmaf(hc[11], k2[z],
                                    fmaf(hc[10], k1[z], y[z])))));
    seirm(yt, beta, alpha, gamma, mu, gm, k6);
#pragma unroll
    for (int z = 0; z < 5; ++z)
        y[z] = fmaf(hc[19], k6[z],
                    fmaf(hc[18], k5[z],
                         fmaf(hc[17], k4[z],
                              fmaf(hc[16], k3[z],
                                   fmaf(hc[15], k1[z], y[z])))));
}

// ---- Projection of 32 per-lane states via V_WMMA_F32_16X16X4_F32 --------
// WMMA computes sum_{z<4} y[row][z]*W[z] for 16 rows per tile (W packed into
// column N=0 of the B operand). The z=4 term and bias are applied per-lane at
// the final coalesced store using the register copy of y[4] (row == lane).
__device__ __forceinline__ void project_store(float* sY, float* sO,
                                              const float y[5], v2f Bm,
                                              float W4, float b0,
                                              float* __restrict__ outp,
                                              int tid) {
#pragma unroll
    for (int z = 0; z < 4; ++z) sY[tid * 4 + z] = y[z];
    __syncthreads();

    const int lane = tid & 31;
    const int m    = lane & 15;
    const int khi  = lane >> 4;   // lanes 0-15: K=0,1 ; lanes 16-31: K=2,3

#pragma unroll
    for (int rb = 0; rb < 2; ++rb) {           // two 16-row tiles -> 32 rows
        const int rowbase = rb * 16;
        v2f A;                                  // 16x4 f32 A-operand striping
        A.x = sY[(rowbase + m) * 4 + (khi ? 2 : 0)];
        A.y = sY[(rowbase + m) * 4 + (khi ? 3 : 1)];
        v8f C = {};
        v8f D = __builtin_amdgcn_wmma_f32_16x16x4_f32(
            false, A, false, Bm, (short)0, C, false, false);
        // Column N=0 lives in lane 0 (M=0..7) and lane 16 (M=8..15).
        if ((lane & 15) == 0) {
#pragma unroll
            for (int r = 0; r < 8; ++r)
                sO[rowbase + khi * 8 + r] = D[r];
        }
    }
    __syncthreads();
    // row == tid: per-lane correction + bias, fully coalesced store.
    outp[tid] = fmaf(y[4], W4, sO[tid]) + b0;
}

__global__ __launch_bounds__(BLOCK) void seirm_dopri5_kernel(
    const float* __restrict__ ze, const float* __restrict__ tarr,
    const float* __restrict__ theta, const float* __restrict__ Wp,
    const float* __restrict__ bp, float* __restrict__ out) {
    __shared__ float sT[TPTS];
    __shared__ float sY[BLOCK * 4];
    __shared__ float sO[BLOCK];

    const int tid   = threadIdx.x;
    const int gbase = blockIdx.x * BLOCK;
    const int b     = gbase + tid;

    for (int i = tid; i < TPTS; i += BLOCK) sT[i] = tarr[i];

    float y[5];
#pragma unroll
    for (int z = 0; z < 5; ++z) y[z] = ze[b * 5 + z];

    const float beta = theta[0], alpha = theta[1], gamma = theta[2],
                mu = theta[3];
    const float gm = gamma + mu;
    const float W0 = Wp[0], W1 = Wp[1], W2 = Wp[2], W3 = Wp[3], W4 = Wp[4];
    const float b0 = bp[0];

    // B operand: W packed into column 0 (N=0) only.
    const int lane = tid & 31;
    v2f Bm;
    Bm.x = (lane == 0) ? W0 : ((lane == 16) ? W2 : 0.0f);
    Bm.y = (lane == 0) ? W1 : ((lane == 16) ? W3 : 0.0f);

    __syncthreads();

    // t = 0: emit projection of the initial state.
    project_store(sY, sO, y, Bm, W4, b0, out + gbase, tid);

    for (int ti = 1; ti < TPTS; ++ti) {
        const float h = (sT[ti] - sT[ti - 1]) * (1.0f / (float)SUB);
        // h-prescaled Butcher tableau (constant across the 8 substeps).
        float hc[20];
        hc[0]  = h * (1.0f / 5.0f);
        hc[1]  = h * (3.0f / 40.0f);
        hc[2]  = h * (9.0f / 40.0f);
        hc[3]  = h * (44.0f / 45.0f);
        hc[4]  = h * (-56.0f / 15.0f);
        hc[5]  = h * (32.0f / 9.0f);
        hc[6]  = h * (19372.0f / 6561.0f);
        hc[7]  = h * (-25360.0f / 2187.0f);
        hc[8]  = h * (64448.0f / 6561.0f);
        hc[9]  = h * (-212.0f / 729.0f);
        hc[10] = h * (9017.0f / 3168.0f);
        hc[11] = h * (-355.0f / 33.0f);
        hc[12] = h * (46732.0f / 5247.0f);
        hc[13] = h * (49.0f / 176.0f);
        hc[14] = h * (-5103.0f / 18656.0f);
        hc[15] = h * (35.0f / 384.0f);
        hc[16] = h * (500.0f / 1113.0f);
        hc[17] = h * (125.0f / 192.0f);
        hc[18] = h * (-2187.0f / 6784.0f);
        hc[19] = h * (11.0f / 84.0f);

        for (int s = 0; s < SUB; ++s)
            dopri5_step(y, hc, beta, alpha, gamma, mu, gm);

        project_store(sY, sO, y, Bm, W4, b0,
                      out + (size_t)ti * BATCH + gbase, tid);
    }
}

extern "C" void kernel_launch(void* const* d_in, const int* in_sizes, int n_in,
                              void* d_out, int out_size, void* d_ws,
                              size_t ws_size, hipStream_t stream) {
    const float* ze    = (const float*)d_in[0];  // [1, B, 5]
    const float* t     = (const float*)d_in[1];  // [T]
    const float* theta = (const float*)d_in[2];  // [4]
    const float* W     = (const float*)d_in[3];  // [1, 5]
    const float* bvec  = (const float*)d_in[4];  // [1]
    float* out         = (float*)d_out;          // [T, B, 1]

    dim3 grid(BATCH / BLOCK);
    dim3 block(BLOCK);
    hipLaunchKernelGGL(seirm_dopri5_kernel, grid, block, 0, stream, ze, t,
                       theta, W, bvec, out);
}